// Transformer_block_70841190580850
// MI455X (gfx1250) — compile-verified
//
#include <hip/hip_runtime.h>
#include <math.h>

// Problem constants (from reference): B=2, M=64, N=64, H=2, C=8, D=4
#define Bz   2
#define Hh   2
#define Cc   8
#define Dd   4
#define SEQ  4096            // 64*64

typedef __attribute__((ext_vector_type(2))) float v2f;
typedef __attribute__((ext_vector_type(8))) float v8f;
typedef __attribute__((ext_vector_type(4))) int   v4i;

__device__ __forceinline__ float gelu_f(float x) {
    // jax.nn.gelu default (approximate=True, tanh form)
    float x3 = x * x * x;
    return 0.5f * x * (1.0f + tanhf(0.7978845608028654f * (x + 0.044715f * x3)));
}

// ---- async global -> LDS copy (CDNA5 GLOBAL_LOAD_ASYNC_TO_LDS_B128) -------
#if defined(__has_builtin)
#if __has_builtin(__builtin_amdgcn_global_load_async_to_lds_b128)
#define USE_ASYNC_LDS 1
#endif
#endif

__device__ __forceinline__ void cp16(const float* __restrict__ g, float* l) {
#ifdef USE_ASYNC_LDS
    // ROCm 7.2 signature: (v4i* global_src, v4i* lds_dst, imm offset, imm cpol)
    __builtin_amdgcn_global_load_async_to_lds_b128((v4i*)g, (v4i*)l, 0, 0);
#else
    *(float4*)l = *(const float4*)g;
#endif
}

__device__ __forceinline__ void cp_fence() {
#ifdef USE_ASYNC_LDS
#if __has_builtin(__builtin_amdgcn_s_wait_asynccnt)
    __builtin_amdgcn_s_wait_asynccnt(0);
#else
    asm volatile("s_wait_asynccnt 0x0" ::: "memory");
#endif
#endif
}

// ---- Stage 1: x1 = circ_conv3x3(gelu(x)) + x ------------------------------
__global__ void k_conv_in(const float* __restrict__ x, const float* __restrict__ W,
                          const float* __restrict__ bias, float* __restrict__ x1) {
    int idx = blockIdx.x * blockDim.x + threadIdx.x;   // (b,p,q,co)
    if (idx >= Bz * SEQ * Cc) return;
    int co  = idx & 7;
    int pix = idx >> 3;
    int q = pix & 63, p = (pix >> 6) & 63, b = pix >> 12;
    float acc = bias[co];
    for (int kh = 0; kh < 3; kh++) {
        int pp = (p + kh + 63) & 63;            // (p+kh-1) mod 64, circular
        for (int kw = 0; kw < 3; kw++) {
            int qq = (q + kw + 63) & 63;
            const float* xp = x + ((b * 64 + pp) * 64 + qq) * Cc;
            const float* wp = W + ((kh * 3 + kw) * Cc) * Cc + co;   // HWIO
            #pragma unroll
            for (int ci = 0; ci < Cc; ci++) acc += gelu_f(xp[ci]) * wp[ci * Cc];
        }
    }
    x1[idx] = acc + x[idx];
}

// ---- Stage 2: y = x1/sqrt(2);  Q/K/V = y @ Wq/Wk/Wv; layout (b,h,i,d) -----
__global__ void k_qkv(const float* __restrict__ x1, const float* __restrict__ Wq,
                      const float* __restrict__ Wk, const float* __restrict__ Wv,
                      float* __restrict__ Q, float* __restrict__ K, float* __restrict__ V) {
    int idx = blockIdx.x * blockDim.x + threadIdx.x;   // (b,i,e)
    if (idx >= Bz * SEQ * Cc) return;
    int e = idx & 7;
    int i = (idx >> 3) & (SEQ - 1);
    int b = idx >> 15;
    const float inv_s2 = 0.70710678118654752f;
    const float* xp = x1 + (((b << 12) + i) * Cc);
    float yq = 0.f, yk = 0.f, yv = 0.f;
    #pragma unroll
    for (int ci = 0; ci < Cc; ci++) {
        float y = xp[ci] * inv_s2;
        yq += y * Wq[ci * 8 + e];
        yk += y * Wk[ci * 8 + e];
        yv += y * Wv[ci * 8 + e];
    }
    int h = e >> 2, dd = e & 3;
    int o = (((b * Hh + h) << 12) + i) * Dd + dd;
    Q[o] = yq; K[o] = yk; V[o] = yv;
}

// ---- Stage 3: fused flash attention, transposed score tiles ---------------
// WMMA:  A = K-tile (16 keys x 4),  B = Q^T (4 x 16 queries)
//   => D[M=key (VGPRs), N=query (lanes)] : each lane owns one query row's
//      8 key-scores -> softmax reduction is a pure in-register max/sum tree.
// Half-waves own disjoint key halves; independent online-softmax states are
// flash-merged once at the end (6 cross-lane ops per wave total).
__global__ void __launch_bounds__(128) k_attn(const float* __restrict__ Q,
                                              const float* __restrict__ K,
                                              const float* __restrict__ V,
                                              const float* __restrict__ pos,
                                              float* __restrict__ A) {
    extern __shared__ float smem[];
    float* sPos = smem;            // 4096 floats  (pos[h] 64x64)
    float* sK   = smem + 4096;     // 16384 floats (K[b,h] 4096x4)
    float* sV   = sK + 16384;      // 16384 floats (V[b,h] 4096x4)

    const int h = blockIdx.y, b = blockIdx.z;
    const int tid = threadIdx.x;
    const float* Kb = K + (size_t)((b * Hh + h) << 12) * Dd;
    const float* Vb = V + (size_t)((b * Hh + h) << 12) * Dd;
    const float* Pb = pos + (h << 12);
    for (int t = tid; t < 1024; t += 128) cp16(Pb + 4 * t, sPos + 4 * t);
    for (int t = tid; t < 4096; t += 128) {
        cp16(Kb + 4 * t, sK + 4 * t);
        cp16(Vb + 4 * t, sV + 4 * t);
    }
    cp_fence();
    __syncthreads();

    const int wave = tid >> 5;
    const int lane = tid & 31;
    const int nIdx = lane & 15;               // query column within tile / A's M
    const int hi   = lane >> 4;               // half-wave: key offset +8, K-dim pair
    const int kd0  = hi ? 2 : 0;
    const int rt   = blockIdx.x * 4 + wave;   // query row tile 0..255
    const int row0 = rt << 4;

    const int iq = row0 + nIdx;               // this lane's query index
    const int ia = iq >> 6, ib = iq & 63;

    // B operand (4x16 f32), fixed: Q^T columns = queries
    const float* Qb = Q + (size_t)((b * Hh + h) << 12) * Dd;
    v2f bq;
    {
        const float2 qq = *(const float2*)&Qb[iq * Dd + kd0];
        bq.x = qq.x; bq.y = qq.y;
    }

    float m = -1e30f, rsum = 0.f;
    float acc0 = 0.f, acc1 = 0.f, acc2 = 0.f, acc3 = 0.f;

    for (int jt = 0; jt < 256; jt++) {
        const int j0 = jt << 4;
        // A operand (16x4 f32): 16 keys of this tile
        v2f ak;
        {
            const float2 kk = *(const float2*)&sK[(j0 + nIdx) * Dd + kd0];
            ak.x = kk.x; ak.y = kk.y;
        }
        v8f c = {};
        c = __builtin_amdgcn_wmma_f32_16x16x4_f32(
                /*neg_a=*/false, ak, /*neg_b=*/false, bq,
                /*c_mod=*/(short)0, c, /*reuse_a=*/false, /*reuse_b=*/false);

        // bias: within a 16-key tile, j>>6 is constant and j&63 never wraps
        const int da64 = ((ia - (j0 >> 6)) & 63) << 6;
        const int dbb  = ib - (j0 & 63);
        const int koff = hi << 3;

        float s[8];
        #pragma unroll
        for (int r = 0; r < 8; r++) {
            const int db = (dbb - (koff + r)) & 63;
            s[r] = (c[r] + sPos[da64 + db]) * 0.5f;   // * 1/sqrt(d), d=4
        }
        const float tm = fmaxf(fmaxf(fmaxf(s[0], s[1]), fmaxf(s[2], s[3])),
                               fmaxf(fmaxf(s[4], s[5]), fmaxf(s[6], s[7])));
        const float mnew = fmaxf(m, tm);
        const float scl  = __expf(m - mnew);
        m = mnew;
        rsum *= scl;
        acc0 *= scl; acc1 *= scl; acc2 *= scl; acc3 *= scl;
        #pragma unroll
        for (int r = 0; r < 8; r++) {
            const float p = __expf(s[r] - mnew);
            rsum += p;
            // V row broadcast: uniform address across the 16-lane half
            const float4 vv = *(const float4*)&sV[(j0 + koff + r) * Dd];
            acc0 += p * vv.x; acc1 += p * vv.y; acc2 += p * vv.z; acc3 += p * vv.w;
        }
    }

    // flash-merge the two half-wave softmax states (same query, disjoint keys)
    const float mo = __shfl_xor(m, 16, 32);
    const float ro = __shfl_xor(rsum, 16, 32);
    const float a0o = __shfl_xor(acc0, 16, 32);
    const float a1o = __shfl_xor(acc1, 16, 32);
    const float a2o = __shfl_xor(acc2, 16, 32);
    const float a3o = __shfl_xor(acc3, 16, 32);
    const float mm  = fmaxf(m, mo);
    const float s1  = __expf(m - mm);
    const float s2  = __expf(mo - mm);
    const float inv = 1.0f / (rsum * s1 + ro * s2);
    if (lane < 16) {
        float4 o;
        o.x = (acc0 * s1 + a0o * s2) * inv;
        o.y = (acc1 * s1 + a1o * s2) * inv;
        o.z = (acc2 * s1 + a2o * s2) * inv;
        o.w = (acc3 * s1 + a3o * s2) * inv;
        *(float4*)&A[(((size_t)(b << 12) + iq) * 8) + h * Dd] = o;
    }
}

// ---- Stage 4: x2 = a @ Wo + x1 --------------------------------------------
__global__ void k_proj(const float* __restrict__ A, const float* __restrict__ Wo,
                       const float* __restrict__ x1, float* __restrict__ x2) {
    int idx = blockIdx.x * blockDim.x + threadIdx.x;   // (b,i,co)
    if (idx >= Bz * SEQ * Cc) return;
    int co = idx & 7, bi = idx >> 3;
    float acc = x1[idx];
    const float* ap = A + bi * 8;
    #pragma unroll
    for (int e = 0; e < 8; e++) acc += ap[e] * Wo[e * 8 + co];
    x2[idx] = acc;
}

// ---- Stage 5: y1 = gelu(conv1x1(x2/sqrt(3), W1) + b1)  (8 -> 32) ----------
__global__ void k_mlp1(const float* __restrict__ x2, const float* __restrict__ W1,
                       const float* __restrict__ b1, float* __restrict__ y1) {
    int idx = blockIdx.x * blockDim.x + threadIdx.x;   // (b,i,oc)
    if (idx >= Bz * SEQ * 32) return;
    int oc = idx & 31, bi = idx >> 5;
    const float inv_s3 = 0.57735026918962576f;
    const float* xp = x2 + bi * 8;
    float acc = b1[oc];
    #pragma unroll
    for (int ci = 0; ci < 8; ci++) acc += (xp[ci] * inv_s3) * W1[ci * 32 + oc];
    y1[idx] = gelu_f(acc);
}

// ---- Stage 6: y2 = gelu(depthwise circ_conv3x3(y1, W2) + b2)  (32 ch) -----
__global__ void k_mlp2(const float* __restrict__ y1, const float* __restrict__ W2,
                       const float* __restrict__ b2, float* __restrict__ y2) {
    int idx = blockIdx.x * blockDim.x + threadIdx.x;   // (b,p,q,ch)
    if (idx >= Bz * SEQ * 32) return;
    int ch = idx & 31;
    int pix = idx >> 5;
    int q = pix & 63, p = (pix >> 6) & 63, b = pix >> 12;
    float acc = b2[ch];
    #pragma unroll
    for (int kh = 0; kh < 3; kh++) {
        int pp = (p + kh + 63) & 63;
        #pragma unroll
        for (int kw = 0; kw < 3; kw++) {
            int qq = (q + kw + 63) & 63;
            acc += y1[((b * 64 + pp) * 64 + qq) * 32 + ch] * W2[(kh * 3 + kw) * 32 + ch];
        }
    }
    y2[idx] = gelu_f(acc);
}

// ---- Stage 7: out = conv1x1(y2, W3) + b3 + x2  (32 -> 8) ------------------
__global__ void k_mlp3(const float* __restrict__ y2, const float* __restrict__ W3,
                       const float* __restrict__ b3, const float* __restrict__ x2,
                       float* __restrict__ out) {
    int idx = blockIdx.x * blockDim.x + threadIdx.x;   // (b,i,co)
    if (idx >= Bz * SEQ * Cc) return;
    int co = idx & 7, bi = idx >> 3;
    float acc = b3[co] + x2[idx];
    const float* yp = y2 + bi * 32;
    #pragma unroll
    for (int ch = 0; ch < 32; ch++) acc += yp[ch] * W3[ch * 8 + co];
    out[idx] = acc;
}

extern "C" void kernel_launch(void* const* d_in, const int* in_sizes, int n_in,
                              void* d_out, int out_size, void* d_ws, size_t ws_size,
                              hipStream_t stream) {
    const float* x      = (const float*)d_in[0];
    const float* conv_W = (const float*)d_in[1];
    const float* conv_b = (const float*)d_in[2];
    const float* Wq     = (const float*)d_in[3];
    const float* Wk     = (const float*)d_in[4];
    const float* Wv     = (const float*)d_in[5];
    const float* pos    = (const float*)d_in[6];
    const float* Wo     = (const float*)d_in[7];
    const float* W1     = (const float*)d_in[8];
    const float* b1     = (const float*)d_in[9];
    const float* W2     = (const float*)d_in[10];
    const float* b2     = (const float*)d_in[11];
    const float* W3     = (const float*)d_in[12];
    const float* b3     = (const float*)d_in[13];

    float* ws  = (float*)d_ws;
    float* x1  = ws;             // 65536
    float* Qd  = ws + 65536;     // 65536  (b,h,i,d)
    float* Kd  = ws + 131072;    // 65536
    float* Vd  = ws + 196608;    // 65536
    float* Aa  = ws + 262144;    // 65536  (b,i,h*d)
    float* x2  = ws + 327680;    // 65536
    float* y1  = ws + 393216;    // 262144 (b,i,32)
    float* y2  = ws + 655360;    // 262144
    float* out = (float*)d_out;

    k_conv_in<<<256, 256, 0, stream>>>(x, conv_W, conv_b, x1);
    k_qkv<<<256, 256, 0, stream>>>(x1, Wq, Wk, Wv, Qd, Kd, Vd);

    // 144 KB dynamic LDS: pos[h] (16KB) + K (64KB) + V (64KB) per workgroup
    dim3 ga(64, Hh, Bz);
    k_attn<<<ga, 128, 147456, stream>>>(Qd, Kd, Vd, pos, Aa);

    k_proj<<<256, 256, 0, stream>>>(Aa, Wo, x1, x2);
    k_mlp1<<<1024, 256, 0, stream>>>(x2, W1, b1, y1);
    k_mlp2<<<1024, 256, 0, stream>>>(y1, W2, b2, y2);
    k_mlp3<<<256, 256, 0, stream>>>(y2, W3, b3, x2, out);
}